// LocalSelfAttention_60962765800209
// MI455X (gfx1250) — compile-verified
//
#include <hip/hip_runtime.h>
#include <stdint.h>

// Problem constants (fixed by reference setup_inputs)
#define NPTS    200000          // B*N
#define NPERB   50000           // N
#define NBATCH  4               // B
#define KK      27              // kernel offsets
#define SS      130             // per-dim encoding stride
#define KEYSPACE (NBATCH*SS*SS*SS)   // 8,788,000
#define LTOT    (NPTS*KK)            // 5,400,000
#define BIGI    0x7fffffff

typedef __attribute__((ext_vector_type(2))) float v2f;
typedef __attribute__((ext_vector_type(8))) float v8f;

// ---- shared helper: stage kernel offsets + linear deltas into LDS ----------
__device__ inline void load_deltas(const int* __restrict__ ko,
                                   int* s_ko, int* s_delta) {
    int t = threadIdx.x;
    if (t < KK * 3) s_ko[t] = ko[t];
    __syncthreads();
    if (t < KK)
        s_delta[t] = (s_ko[3*t + 0] * SS + s_ko[3*t + 1]) * SS + s_ko[3*t + 2];
    __syncthreads();
}

// ---- 1: init first-occurrence table and key-tensor output ------------------
__global__ void k_init(int* __restrict__ first_occ, int* __restrict__ keys_out) {
    int i = blockIdx.x * blockDim.x + threadIdx.x;
    if (i < KEYSPACE)  first_occ[i] = BIGI;
    if (i < 3 * LTOT)  keys_out[i]  = -1;
}

// ---- 2: atomicMin of flat index into direct-mapped key table ---------------
__global__ void k_scatter(const int* __restrict__ coords,
                          const int* __restrict__ bidx,
                          const int* __restrict__ ko,
                          int* __restrict__ first_occ,
                          int* __restrict__ basebuf) {
    __shared__ int s_ko[KK * 3];
    __shared__ int s_delta[KK];
    load_deltas(ko, s_ko, s_delta);

    int p = blockIdx.x * 256 + threadIdx.x;
    if (p >= NPTS) return;
    __builtin_prefetch(&coords[3 * p], 0, 0);     // global_prefetch_b8
    int b = bidx[p];
    int x = coords[3*p + 0], y = coords[3*p + 1], z = coords[3*p + 2];
    int base = ((b * SS + x + 1) * SS + y + 1) * SS + z + 1;
    basebuf[p] = base;
    int flat0 = p * KK;
#pragma unroll
    for (int k = 0; k < KK; ++k)
        atomicMin(&first_occ[base + s_delta[k]], flat0 + k);
}

// ---- 3: winner bitmask per point + per-block population counts -------------
__global__ void k_flags(const int* __restrict__ basebuf,
                        const int* __restrict__ ko,
                        const int* __restrict__ first_occ,
                        int* __restrict__ flagsbuf,
                        int* __restrict__ partials) {
    __shared__ int s_ko[KK * 3];
    __shared__ int s_delta[KK];
    __shared__ int red[256];
    load_deltas(ko, s_ko, s_delta);

    int t = threadIdx.x;
    int p = blockIdx.x * 256 + t;
    int flags = 0, tsum = 0;
    if (p < NPTS) {
        int base  = basebuf[p];
        int flat0 = p * KK;
#pragma unroll
        for (int k = 0; k < KK; ++k) {
            if (first_occ[base + s_delta[k]] == flat0 + k) { flags |= (1 << k); ++tsum; }
        }
        flagsbuf[p] = flags;
    }
    red[t] = tsum;
    __syncthreads();
    for (int s = 128; s > 0; s >>= 1) {
        if (t < s) red[t] += red[t + s];
        __syncthreads();
    }
    if (t == 0) partials[blockIdx.x] = red[0];
}

// ---- 4: exclusive scan of 782 block counts + num_unique --------------------
__global__ void k_scanpart(const int* __restrict__ partials,
                           int* __restrict__ scanned,
                           int* __restrict__ out_nu, int nblocks) {
    if (blockIdx.x == 0 && threadIdx.x == 0) {
        int acc = 0;
        for (int i = 0; i < nblocks; ++i) { scanned[i] = acc; acc += partials[i]; }
        *out_nu = acc;
    }
}

// ---- 5: WMMA matrix-method block prefix scan -> ranks + key tensor ---------
// 256 thread-sums viewed as 16x16 f32 matrix X (row-major over tid).
// Row-inclusive prefixes: P = X * U (U upper-triangular ones), computed as
// four chained v_wmma_f32_16x16x4_f32 accumulations over K-chunks of 4.
// All values are small integers (<= 6912), exact in f32.
__global__ void k_rank(const int* __restrict__ basebuf,
                       const int* __restrict__ flagsbuf,
                       const int* __restrict__ ko,
                       const int* __restrict__ scanned,
                       int* __restrict__ rank_tab,
                       int* __restrict__ keys_out) {
    __shared__ int   s_ko[KK * 3];
    __shared__ int   s_delta[KK];
    __shared__ float s_sums[256];
    __shared__ float s_scan[256];
    __shared__ float s_rowoff[16];
    load_deltas(ko, s_ko, s_delta);

    int t = threadIdx.x;
    int p = blockIdx.x * 256 + t;
    int flags = 0, base = 0;
    if (p < NPTS) { flags = flagsbuf[p]; base = basebuf[p]; }
    int tsum = __popc(flags);
    s_sums[t] = (float)tsum;
    __syncthreads();

    if (t < 32) {                      // wave 0 only: EXEC all-ones within wave
        int m = t & 15;                // lane's row of A / column index n of P
        int h = t >> 4;                // lane half
        v8f acc = {};
#pragma unroll
        for (int k = 0; k < 4; ++k) {
            // A 16x4: vgpr v holds K = 4k + v + 2h for row m
            v2f a;
            a.x = s_sums[m * 16 + 4 * k + 2 * h + 0];
            a.y = s_sums[m * 16 + 4 * k + 2 * h + 1];
            // B 4x16: U rows 4k..4k+3; vgpr v holds row (4k + v + 2h), col = m
            v2f bm;
            bm.x = ((4 * k + 2 * h + 0) <= m) ? 1.0f : 0.0f;
            bm.y = ((4 * k + 2 * h + 1) <= m) ? 1.0f : 0.0f;
            acc = __builtin_amdgcn_wmma_f32_16x16x4_f32(
                false, a, false, bm, (short)0, acc, false, false);
        }
        // C/D layout: vgpr v holds P row (v + 8h), column = m
#pragma unroll
        for (int v = 0; v < 8; ++v)
            s_scan[(v + 8 * h) * 16 + m] = acc[v];
    }
    __syncthreads();

    if (t == 0) {                      // 16-row offset scan (trivial)
        float r = 0.f;
#pragma unroll
        for (int i = 0; i < 16; ++i) { s_rowoff[i] = r; r += s_scan[i * 16 + 15]; }
    }
    __syncthreads();

    float incl = s_scan[t] + s_rowoff[t >> 4];
    int   excl = (int)incl - tsum;     // exclusive in-block prefix (exact)

    if (p < NPTS && flags) {
        int rank = scanned[blockIdx.x] + excl;
        int bx = (base / (SS * SS)) % SS - 1;
        int by = (base / SS) % SS - 1;
        int bz =  base % SS - 1;
#pragma unroll
        for (int k = 0; k < KK; ++k) {
            if ((flags >> k) & 1) {
                rank_tab[base + s_delta[k]] = rank;
                keys_out[3 * rank + 0] = bx + s_ko[3 * k + 0];
                keys_out[3 * rank + 1] = by + s_ko[3 * k + 1];
                keys_out[3 * rank + 2] = bz + s_ko[3 * k + 2];
                ++rank;
            }
        }
    }
}

// ---- 6: emit index maps (coalesced over flat L) ----------------------------
__global__ void k_emit(const int* __restrict__ basebuf,
                       const int* __restrict__ ko,
                       const int* __restrict__ rank_tab,
                       int* __restrict__ out_in,
                       int* __restrict__ out_oidx,
                       int* __restrict__ out_rel) {
    __shared__ int s_ko[KK * 3];
    __shared__ int s_delta[KK];
    load_deltas(ko, s_ko, s_delta);

    int i = blockIdx.x * 256 + threadIdx.x;
    if (i >= LTOT) return;
    int p = i / KK;
    int k = i - p * KK;
    out_in[i]   = p % NPERB;                       // n
    out_rel[i]  = k;                               // kernel index
    out_oidx[i] = rank_tab[basebuf[p] + s_delta[k]];
}

// ---------------------------------------------------------------------------
extern "C" void kernel_launch(void* const* d_in, const int* in_sizes, int n_in,
                              void* d_out, int out_size, void* d_ws, size_t ws_size,
                              hipStream_t stream) {
    const int* coords = (const int*)d_in[0];   // [B,N,3] int32
    const int* bidx   = (const int*)d_in[1];   // [B,N]   int32
    const int* ko     = (const int*)d_in[2];   // [27,3]  int32

    int* out      = (int*)d_out;               // int32 outputs, concatenated
    int* out_in   = out;                       // [B,N,K]
    int* out_oidx = out + LTOT;                // [B,N,K]
    int* out_rel  = out + 2 * LTOT;            // [B,N,K]
    int* keys_out = out + 3 * LTOT;            // [L,3]
    int* out_nu   = out + 6 * LTOT;            // scalar

    char* ws = (char*)d_ws;
    int* first_occ = (int*)ws;  ws += (size_t)KEYSPACE * 4;
    int* rank_tab  = (int*)ws;  ws += (size_t)KEYSPACE * 4;
    int* basebuf   = (int*)ws;  ws += (size_t)NPTS * 4;
    int* flagsbuf  = (int*)ws;  ws += (size_t)NPTS * 4;
    const int nbp = (NPTS + 255) / 256;        // 782 point-blocks
    int* partials  = (int*)ws;  ws += (size_t)(nbp + 1) * 4;
    int* scanned   = (int*)ws;

    int ni = (KEYSPACE > 3 * LTOT) ? KEYSPACE : 3 * LTOT;
    k_init    <<<(ni + 255) / 256, 256, 0, stream>>>(first_occ, keys_out);
    k_scatter <<<nbp,             256, 0, stream>>>(coords, bidx, ko, first_occ, basebuf);
    k_flags   <<<nbp,             256, 0, stream>>>(basebuf, ko, first_occ, flagsbuf, partials);
    k_scanpart<<<1,                 1, 0, stream>>>(partials, scanned, out_nu, nbp);
    k_rank    <<<nbp,             256, 0, stream>>>(basebuf, flagsbuf, ko, scanned, rank_tab, keys_out);
    k_emit    <<<(LTOT + 255) / 256, 256, 0, stream>>>(basebuf, ko, rank_tab, out_in, out_oidx, out_rel);
}